// SimpleNet_87720412054077
// MI455X (gfx1250) — compile-verified
//
#include <hip/hip_runtime.h>

typedef float v2f __attribute__((ext_vector_type(2)));
typedef float v8f __attribute__((ext_vector_type(8)));

#define DIM    8
#define HIDN   32
#define NIDS   48
#define NCOEF  6
#define CTAB   48      // DIM * NCOEF
#define ROWPAD 52      // padded LDS row stride (floats) to spread banks

// ---------------------------------------------------------------------------
// Kernel A (1 wave): table[id][n] = scale_sp[n/6] *
//        ( relu(emb[id] @ W1^T + b1) @ W2^T + b2 )[n]
// Implemented as two fp32 WMMA GEMMs: (48x8)@(8x32), then (48x32)@(32x48).
// ---------------------------------------------------------------------------
__global__ __launch_bounds__(32) void precompute_coef(
    const float* __restrict__ emb,       // (48,8)
    const float* __restrict__ W1,        // (32,8)
    const float* __restrict__ b1,        // (32)
    const float* __restrict__ W2,        // (48,32)
    const float* __restrict__ b2,        // (48)
    const float* __restrict__ scale_sp,  // (8)
    float* __restrict__ table)           // (48,48) scratch
{
    __shared__ float sE[NIDS * DIM];
    __shared__ float sW1[HIDN * DIM];
    __shared__ float sW2[CTAB * HIDN];
    __shared__ float sH[NIDS * HIDN];
    __shared__ float sB1[HIDN];
    __shared__ float sB2[CTAB];
    __shared__ float sS[DIM];

    const int tid = threadIdx.x;           // 0..31 (wave32)
    for (int i = tid; i < NIDS * DIM;  i += 32) sE[i]  = emb[i];
    for (int i = tid; i < HIDN * DIM;  i += 32) sW1[i] = W1[i];
    for (int i = tid; i < CTAB * HIDN; i += 32) sW2[i] = W2[i];
    for (int i = tid; i < CTAB;        i += 32) sB2[i] = b2[i];
    sB1[tid] = b1[tid];
    if (tid < DIM) sS[tid] = scale_sp[tid];
    __syncthreads();

    const int half = tid >> 4;             // which 16-lane group
    const int lm   = tid & 15;             // M index for A-frag, N index for B/C

    // ---- GEMM1: H = relu(E @ W1^T + b1)   M=48, N=32, K=8 ----
    for (int mt = 0; mt < 3; ++mt) {
        for (int nt = 0; nt < 2; ++nt) {
            v8f acc = {};
            #pragma unroll
            for (int k0 = 0; k0 < DIM; k0 += 4) {
                const int ka = k0 + 2 * half;      // VGPR0: K={0,2}, VGPR1: K={1,3}
                v2f a, b;
                a[0] = sE[(mt * 16 + lm) * DIM + ka];
                a[1] = sE[(mt * 16 + lm) * DIM + ka + 1];
                // B = W1^T : B[k][n] = W1[n][k]
                b[0] = sW1[(nt * 16 + lm) * DIM + ka];
                b[1] = sW1[(nt * 16 + lm) * DIM + ka + 1];
                acc = __builtin_amdgcn_wmma_f32_16x16x4_f32(
                        false, a, false, b, (short)0, acc, false, false);
            }
            const int col = nt * 16 + lm;
            #pragma unroll
            for (int r = 0; r < 8; ++r) {          // C VGPR r: M = r + 8*half
                const int row = mt * 16 + r + 8 * half;
                float v = acc[r] + sB1[col];
                sH[row * HIDN + col] = v > 0.0f ? v : 0.0f;
            }
        }
    }
    __syncthreads();

    // ---- GEMM2: C = H @ W2^T + b2, fused scale_sp   M=48, N=48, K=32 ----
    for (int mt = 0; mt < 3; ++mt) {
        for (int nt = 0; nt < 3; ++nt) {
            v8f acc = {};
            #pragma unroll
            for (int k0 = 0; k0 < HIDN; k0 += 4) {
                const int ka = k0 + 2 * half;
                v2f a, b;
                a[0] = sH[(mt * 16 + lm) * HIDN + ka];
                a[1] = sH[(mt * 16 + lm) * HIDN + ka + 1];
                // B = W2^T : B[k][n] = W2[n][k]
                b[0] = sW2[(nt * 16 + lm) * HIDN + ka];
                b[1] = sW2[(nt * 16 + lm) * HIDN + ka + 1];
                acc = __builtin_amdgcn_wmma_f32_16x16x4_f32(
                        false, a, false, b, (short)0, acc, false, false);
            }
            const int col = nt * 16 + lm;          // n = d*6 + j
            const float s = sS[col / NCOEF];
            #pragma unroll
            for (int r = 0; r < 8; ++r) {
                const int row = mt * 16 + r + 8 * half;   // id
                table[row * CTAB + col] = (acc[r] + sB2[col]) * s;
            }
        }
    }
}

// ---------------------------------------------------------------------------
// Kernel B: bandwidth-bound main pass. 40 B of HBM traffic per element.
// out[b] = scale_base * sum_d silu(budget[b,d])
//        + sum_d sum_{j in 4 nonzero} N_j(budget[b,d]) * table[x[b]][d*6+j]
// Uniform-knot closed-form cubic basis (both intervals are fully interior).
// ---------------------------------------------------------------------------
__global__ __launch_bounds__(256) void kan_forward(
    const int*   __restrict__ x,          // (B)
    const float* __restrict__ budget,     // (B,8)
    const float* __restrict__ grid,       // (8,10)
    const float* __restrict__ scale_base, // (1)
    const float* __restrict__ table,      // (48,48) from kernel A
    float* __restrict__ out, int n)
{
    __shared__ float sT[NIDS * ROWPAD];
    for (int i = threadIdx.x; i < NIDS * CTAB; i += blockDim.x)
        sT[(i / CTAB) * ROWPAD + (i % CTAB)] = table[i];
    __syncthreads();

    const float g4    = grid[4];
    const float g5    = grid[5];
    const float inv_h = 1.0f / (g5 - g4);
    const float sb    = scale_base[0];

    const int stride = gridDim.x * blockDim.x;
    for (int b = blockIdx.x * blockDim.x + threadIdx.x; b < n; b += stride) {
        const int id = x[b];
        const float4 q0 = ((const float4*)(budget + (size_t)b * DIM))[0];
        const float4 q1 = ((const float4*)(budget + (size_t)b * DIM))[1];
        const float t[8] = { q0.x, q0.y, q0.z, q0.w, q1.x, q1.y, q1.z, q1.w };

        const float* row = &sT[id * ROWPAD];
        float acc = 0.0f, bacc = 0.0f;
        #pragma unroll
        for (int d = 0; d < DIM; ++d) {
            const float v = t[d];
            bacc += v * (1.0f / (1.0f + __expf(-v)));          // silu

            const int   i5 = (v >= g5) ? 1 : 0;                // interval 4 or 5
            const float u  = (v - (i5 ? g5 : g4)) * inv_h;     // u in [0,1)
            const float um = 1.0f - u;
            const float u2 = u * u, u3 = u2 * u;
            const float n0 = um * um * um * (1.0f / 6.0f);
            const float n1 = ( 3.0f * u3 - 6.0f * u2 + 4.0f) * (1.0f / 6.0f);
            const float n2 = (-3.0f * u3 + 3.0f * u2 + 3.0f * u + 1.0f) * (1.0f / 6.0f);
            const float n3 = u3 * (1.0f / 6.0f);

            const float* c = row + d * NCOEF + 1 + i5;         // j0 = i-3
            acc = fmaf(n0, c[0], acc);
            acc = fmaf(n1, c[1], acc);
            acc = fmaf(n2, c[2], acc);
            acc = fmaf(n3, c[3], acc);
        }
        out[b] = sb * bacc + acc;
    }
}

// ---------------------------------------------------------------------------
extern "C" void kernel_launch(void* const* d_in, const int* in_sizes, int n_in,
                              void* d_out, int out_size, void* d_ws, size_t ws_size,
                              hipStream_t stream) {
    (void)in_sizes; (void)n_in; (void)ws_size;
    const int*   x          = (const int*)  d_in[0];
    const float* budget     = (const float*)d_in[1];
    const float* emb        = (const float*)d_in[2];
    const float* W1         = (const float*)d_in[3];
    const float* b1         = (const float*)d_in[4];
    const float* W2         = (const float*)d_in[5];
    const float* b2         = (const float*)d_in[6];
    const float* grid       = (const float*)d_in[7];
    const float* scale_base = (const float*)d_in[8];
    const float* scale_sp   = (const float*)d_in[9];
    float* out   = (float*)d_out;
    float* table = (float*)d_ws;   // 48*48*4 = 9216 bytes

    precompute_coef<<<1, 32, 0, stream>>>(emb, W1, b1, W2, b2, scale_sp, table);

    const int n = out_size;                 // B = 2,097,152
    const int blocks = 2048;                // grid-stride; ~4 elems/thread
    kan_forward<<<blocks, 256, 0, stream>>>(x, budget, grid, scale_base, table, out, n);
}